// SpanRanking_72249939853626
// MI455X (gfx1250) — compile-verified
//
#include <hip/hip_runtime.h>
#include <hip/hip_bf16.h>
#include <math.h>

#define BB 32
#define SS 512
#define HH 1024
#define MM 8
#define NTOK (BB * SS)          // 16384
#define NKEYS (NTOK * MM)       // 131072
#define TOPK 4915               // int(0.3 * B * S)
#define LN2F 0.6931471805599453f
#define NBPK ((HH / 2) * 16)    // packed-B entries (float2 each) = 8192

typedef float v2f __attribute__((ext_vector_type(2)));
typedef float v8f __attribute__((ext_vector_type(8)));

// ---------------- Kernel 1: query = w_in @ term_weight + b_in ----------------
__global__ __launch_bounds__(256) void k_query(const float* __restrict__ w_in,
                                               const float* __restrict__ tw,
                                               const float* __restrict__ b_in,
                                               float* __restrict__ q) {
  int row  = blockIdx.x * 8 + (threadIdx.x >> 5);   // 4 blocks * 8 waves = 32 rows
  int lane = threadIdx.x & 31;
  const float* wr = w_in + (size_t)row * HH;
  float s = 0.f;
  for (int j = lane; j < HH; j += 32) s += wr[j] * tw[j];
  for (int off = 16; off; off >>= 1) s += __shfl_xor(s, off, 32);
  if (lane == 0) q[row] = s + b_in[row];
}

// ------- Kernel 1b: pack B operand for WMMA --------------------------------
// bpk[(k/2)*16 + n] = { B[k][n], B[k+1][n] }, B[:,0]=query, B[:,1]=score_w, else 0
__global__ __launch_bounds__(256) void k_packB(const float* __restrict__ q,
                                               const float* __restrict__ sw,
                                               v2f* __restrict__ bpk) {
  int i = blockIdx.x * 256 + threadIdx.x;           // 0..NBPK-1
  int k = (i >> 4) * 2;
  int n = i & 15;
  v2f v;
  v.x = (n == 0) ? q[k]     : (n == 1) ? sw[k]     : 0.f;
  v.y = (n == 0) ? q[k + 1] : (n == 1) ? sw[k + 1] : 0.f;
  bpk[i] = v;
}

// ------- Kernel 2: d[t]=hid[t]·q, e[t]=hid[t]·sw via V_WMMA_F32_16X16X4_F32 -------
// A (16x4 f32): lanes 0-15 rows M=0..15 hold K={k0,k0+1}; lanes 16-31 hold K={k0+2,k0+3}
// B (4x16 f32): col N=lane&15; loaded branchlessly from packed table
// D (16x16 f32, v8f): lanes 0-15 -> M=r, lanes 16-31 -> M=8+r, N=lane&15
__global__ __launch_bounds__(256) void k_proj_wmma(const float* __restrict__ hid,
                                                   const v2f* __restrict__ bpk,
                                                   float* __restrict__ dvec,
                                                   float* __restrict__ evec) {
  int wave = blockIdx.x * (blockDim.x >> 5) + (threadIdx.x >> 5);
  int lane = threadIdx.x & 31;
  int n    = lane & 15;        // A row (M) and B col (N)
  int half = lane >> 4;        // selects K pair
  int base = wave * 16;        // first token of this 16-token tile
  const float* ap = hid + (size_t)(base + n) * HH + 2 * half;
  const v2f*   bp = bpk + half * 16 + n;

  v8f acc = {};
  for (int k0 = 0; k0 < HH; k0 += 4) {
    v2f a = *(const v2f*)(ap + k0);   // global_load_b64, stream of hidden_states
    v2f b = bp[k0 * 8];               // global_load_b64, (k0/2)*16 entries, L2-resident
    acc = __builtin_amdgcn_wmma_f32_16x16x4_f32(false, a, false, b,
                                                (short)0, acc, false, false);
  }
  if (n < 2) {
    float* out = (n == 0) ? dvec : evec;
#pragma unroll
    for (int r = 0; r < 8; ++r) out[base + half * 8 + r] = acc[r];
  }
}

// ------- Kernel 3: per-(b,s) softmax spans -> keys, gold flags, loss partials -------
__device__ inline float softplusf(float z) {  // log(1+exp(z)), stable
  return (z > 0.f) ? z + log1pf(expf(-z)) : log1pf(expf(z));
}

__global__ __launch_bounds__(256) void k_scores(const float* __restrict__ dvec,
                                                const float* __restrict__ evec,
                                                const int* __restrict__ gold_mask,
                                                const int* __restrict__ seq_len,
                                                const float* __restrict__ score_b_p,
                                                unsigned* __restrict__ keys,
                                                int* __restrict__ goldf,
                                                float* __restrict__ blk_loss,
                                                int* __restrict__ blk_nv) {
  int t = blockIdx.x * blockDim.x + threadIdx.x;   // b*S + s
  int b = t / SS, s = t - b * SS;
  float sb = score_b_p[0];
  int len = seq_len[b];

  float dd[MM], ee[MM];
#pragma unroll
  for (int m = 0; m < MM; ++m) {
    int p = s + m; p = (p < SS - 1) ? p : SS - 1;
    dd[m] = dvec[b * SS + p];
    ee[m] = evec[b * SS + p];
  }

  float lsum = 0.f; int nv = 0;
#pragma unroll
  for (int l = 0; l < MM; ++l) {
    bool valid = (s + l + 1) <= len;
    float mx = dd[0];
    for (int m = 1; m <= l; ++m) mx = fmaxf(mx, dd[m]);
    float se = 0.f, we = 0.f;
    for (int m = 0; m <= l; ++m) {
      float w = expf(dd[m] - mx);
      se += w; we += w * ee[m];
    }
    float score = we / se + sb;
    bool gold = (gold_mask[t * MM + l] == 0) && valid;
    unsigned key = 0u;
    if (valid) {
      unsigned u = __float_as_uint(score);
      key = (u & 0x80000000u) ? ~u : (u | 0x80000000u);
      nv++;
      lsum += gold ? softplusf(-score) : LN2F;
    }
    keys[t * MM + l]  = key;
    goldf[t * MM + l] = gold ? 1 : 0;
  }

  __shared__ float sf[256];
  __shared__ int   si[256];
  int tid = threadIdx.x;
  sf[tid] = lsum; si[tid] = nv;
  __syncthreads();
  for (int off = 128; off; off >>= 1) {
    if (tid < off) { sf[tid] += sf[tid + off]; si[tid] += si[tid + off]; }
    __syncthreads();
  }
  if (tid == 0) { blk_loss[blockIdx.x] = sf[0]; blk_nv[blockIdx.x] = si[0]; }
}

// ------- Kernel 4: radix-select K-th largest, precision + loss finalize -------
__global__ __launch_bounds__(1024) void k_finalize(const unsigned* __restrict__ keys,
                                                   const int* __restrict__ goldf,
                                                   const float* __restrict__ blk_loss,
                                                   const int* __restrict__ blk_nv,
                                                   int nblk, float* __restrict__ out) {
  __shared__ unsigned hist[256];
  __shared__ unsigned s_prefix;
  __shared__ int s_remaining;
  __shared__ int ired[1024];
  __shared__ int tot_gt_gold, tot_eq, tot_eq_gold;
  int tid = threadIdx.x;
  if (tid == 0) { s_prefix = 0u; s_remaining = TOPK; }
  __syncthreads();

  for (int pass = 0; pass < 4; ++pass) {
    int shift = 24 - 8 * pass;
    if (tid < 256) hist[tid] = 0u;
    __syncthreads();
    unsigned pre = s_prefix;
    for (int i = tid; i < NKEYS; i += 1024) {
      unsigned k = keys[i];
      unsigned hi = (unsigned)(((unsigned long long)k) >> (shift + 8));  // ok at 32
      if (hi == pre) atomicAdd(&hist[(k >> shift) & 255u], 1u);
    }
    __syncthreads();
    if (tid == 0) {
      int rem = s_remaining;
      int d = 255;
      for (; d > 0; --d) {
        int c = (int)hist[d];
        if (rem <= c) break;
        rem -= c;
      }
      s_prefix = (pre << 8) | (unsigned)d;
      s_remaining = rem;
    }
    __syncthreads();
  }
  unsigned tau = s_prefix;
  int need = s_remaining;

  int gt_gold = 0, eq_tot = 0, eq_gold = 0;
  for (int i = tid; i < NKEYS; i += 1024) {
    unsigned k = keys[i];
    if (k > tau) gt_gold += goldf[i];
    else if (k == tau) { eq_tot++; eq_gold += goldf[i]; }
  }
  ired[tid] = gt_gold; __syncthreads();
  for (int off = 512; off; off >>= 1) { if (tid < off) ired[tid] += ired[tid + off]; __syncthreads(); }
  if (tid == 0) tot_gt_gold = ired[0]; __syncthreads();
  ired[tid] = eq_tot; __syncthreads();
  for (int off = 512; off; off >>= 1) { if (tid < off) ired[tid] += ired[tid + off]; __syncthreads(); }
  if (tid == 0) tot_eq = ired[0]; __syncthreads();
  ired[tid] = eq_gold; __syncthreads();
  for (int off = 512; off; off >>= 1) { if (tid < off) ired[tid] += ired[tid + off]; __syncthreads(); }
  if (tid == 0) tot_eq_gold = ired[0]; __syncthreads();

  if (tid == 0) {
    int right;
    if (tot_eq == need) {
      right = tot_gt_gold + tot_eq_gold;           // common case: all ties included
    } else {                                       // rare: ties broken by lowest index
      right = tot_gt_gold;
      int got = 0;
      for (int i = 0; i < NKEYS && got < need; ++i)
        if (keys[i] == tau) { right += goldf[i]; ++got; }
    }
    float lsum = 0.f; int nv = 0;
    for (int i = 0; i < nblk; ++i) { lsum += blk_loss[i]; nv += blk_nv[i]; }
    out[0] = lsum / (float)nv;
    out[1] = (float)right / (float)TOPK;
  }
}

// ---------------------------------- launcher ----------------------------------
extern "C" void kernel_launch(void* const* d_in, const int* in_sizes, int n_in,
                              void* d_out, int out_size, void* d_ws, size_t ws_size,
                              hipStream_t stream) {
  const float* hidden  = (const float*)d_in[0];
  const float* tw      = (const float*)d_in[1];
  const float* w_in    = (const float*)d_in[2];
  const float* b_in    = (const float*)d_in[3];
  const float* score_w = (const float*)d_in[4];
  const float* score_b = (const float*)d_in[5];
  const int*   seq_len = (const int*)d_in[6];
  const int*   gmask   = (const int*)d_in[7];
  float* out = (float*)d_out;

  char* ws = (char*)d_ws;
  size_t off = 0;
  float* q      = (float*)(ws + off); off += HH * sizeof(float);
  float* dvec   = (float*)(ws + off); off += NTOK * sizeof(float);
  float* evec   = (float*)(ws + off); off += NTOK * sizeof(float);
  unsigned* key = (unsigned*)(ws + off); off += NKEYS * sizeof(unsigned);
  int* goldf    = (int*)(ws + off); off += NKEYS * sizeof(int);
  float* bloss  = (float*)(ws + off); off += 64 * sizeof(float);
  int* bnv      = (int*)(ws + off); off += 64 * sizeof(int);
  v2f* bpk      = (v2f*)(ws + off); off += NBPK * sizeof(v2f);
  (void)ws_size; (void)in_sizes; (void)n_in; (void)out_size;

  k_query<<<HH / 8 / 32, 256, 0, stream>>>(w_in, tw, b_in, q);            // 4 blocks
  k_packB<<<NBPK / 256, 256, 0, stream>>>(q, score_w, bpk);               // 32 blocks
  k_proj_wmma<<<NTOK / 16 / 8, 256, 0, stream>>>(hidden, bpk, dvec, evec); // 128 blocks
  k_scores<<<NTOK / 256, 256, 0, stream>>>(dvec, evec, gmask, seq_len, score_b,
                                           key, goldf, bloss, bnv);       // 64 blocks
  k_finalize<<<1, 1024, 0, stream>>>(key, goldf, bloss, bnv, NTOK / 256, out);
}